// NPLinear_31421980738253
// MI455X (gfx1250) — compile-verified
//
#include <hip/hip_runtime.h>

typedef float v2f __attribute__((ext_vector_type(2)));
typedef float v8f __attribute__((ext_vector_type(8)));
typedef int   v4i __attribute__((ext_vector_type(4)));

#define AS1 __attribute__((address_space(1)))
#define AS3 __attribute__((address_space(3)))

#if defined(__has_builtin)
#if __has_builtin(__builtin_amdgcn_global_load_async_to_lds_b128)
#define HAVE_ASYNC_LDS 1
#endif
#endif

struct PtrPack {
  const float* w[3];
  const float* bias[3];
  const float* lW[3];  const float* lb[3];
  const float* rcW[3]; const float* rcb[3];
  const float* rW[3];  const float* rb[3];
  const float* cW[3];  const float* cb[3];
  const float* bW[3];  const float* bb[3];
  const float* brcW[3];const float* brcb[3];
  float* ws;
  float* out;
};

// ---- dimensions ----
__device__ __forceinline__ int odim_of(int l){ return l==2 ? 64 : 256; }
__device__ __forceinline__ int ndim_of(int l){ return l==0 ? 128 : 256; }

// ---- workspace offsets (in floats) ----
__device__ __forceinline__ long long rm_off(int l){ return l==0?0LL:(l==1?65536LL:196608LL); }
__device__ __forceinline__ long long cm_off(int l){ return l==0?327680LL:(l==1?458752LL:589824LL); }
#define RCM_OFF 622592LL
#define BM_OFF  624128LL
#define WB_OFF  625664LL
#define ZC_OFF  628736LL
__device__ __forceinline__ long long z3_off(int l){ return l==0?630272LL:(l==1?695808LL:826880LL); }
__device__ __forceinline__ long long z4_off(int l){ return l==0?957952LL:(l==1?1089024LL:1220096LL); }
// ---- output offsets (in floats) ----
__device__ __forceinline__ long long ow_off(int l){ return l==0?0LL:(l==1?16908288LL:50593792LL); }
__device__ __forceinline__ long long ob_off(int l){ return l==0?16777216LL:(l==1?50462720LL:58982400LL); }

__device__ __forceinline__ float wave_sum32(float v) {
  #pragma unroll
  for (int off = 16; off > 0; off >>= 1) v += __shfl_xor(v, off, 32);
  return v;
}

// ================= Pass 1a: per-(layer,b,c) row/col/total means of w =================
__global__ __launch_bounds__(256) void k_reduce_w(PtrPack P) {
  int blk = blockIdx.x;
  int layer = blk >> 9;        // 512 (b,c) planes per layer
  int bc = blk & 511;
  const int o = odim_of(layer), n = ndim_of(layer);
  const float* base = P.w[layer] + (long long)bc * o * n;
  __shared__ float scol[256];
  __shared__ float stot;
  const int tid = threadIdx.x;
  scol[tid] = 0.f;
  if (tid == 0) stot = 0.f;
  __syncthreads();
  const int wave = tid >> 5, lane = tid & 31;
  float acc[8] = {0,0,0,0,0,0,0,0};
  const int nseg = n >> 7;
  for (int row = wave; row < o; row += 8) {
    float rsum = 0.f;
    for (int seg = 0; seg < nseg; ++seg) {
      float4 v = *(const float4*)(base + (long long)row*n + seg*128 + 4*lane);
      acc[seg*4+0]+=v.x; acc[seg*4+1]+=v.y; acc[seg*4+2]+=v.z; acc[seg*4+3]+=v.w;
      rsum += (v.x+v.y) + (v.z+v.w);
    }
    rsum = wave_sum32(rsum);
    if (lane == 0) {
      P.ws[cm_off(layer) + (long long)bc*o + row] = rsum * (1.0f/(float)n);
      atomicAdd(&stot, rsum);
    }
  }
  for (int seg = 0; seg < nseg; ++seg)
    #pragma unroll
    for (int k = 0; k < 4; ++k)
      atomicAdd(&scol[seg*128 + 4*lane + k], acc[seg*4+k]);
  __syncthreads();
  if (tid < n) P.ws[rm_off(layer) + (long long)bc*n + tid] = scol[tid] * (1.0f/(float)o);
  if (tid == 0) P.ws[RCM_OFF + layer*512 + bc] = stot * (1.0f/(float)(o*n));
}

// ================= Pass 1b: bias means bm[layer][b,c] =================
__global__ __launch_bounds__(256) void k_reduce_b(PtrPack P) {
  const int layer = blockIdx.x / 64;
  const int b = blockIdx.x % 64;
  const int o = odim_of(layer);
  const int wave = threadIdx.x >> 5, lane = threadIdx.x & 31;   // wave == channel
  const float* src = P.bias[layer] + ((long long)(b*8 + wave)) * o;
  float s = 0.f;
  for (int k = lane; k < o; k += 32) s += src[k];
  s = wave_sum32(s);
  if (lane == 0) P.ws[BM_OFF + layer*512 + b*8 + wave] = s * (1.0f/(float)o);
}

// ================= wb = concat(rcm0,rcm1,rcm2,bm0,bm1,bm2)  (B,48) =================
__global__ __launch_bounds__(256) void k_wb(PtrPack P) {
  int idx = blockIdx.x*256 + threadIdx.x;          // 3072 total
  if (idx >= 3072) return;
  const int b = idx / 48, j = idx % 48;
  float v;
  if (j < 24) v = P.ws[RCM_OFF + (long long)(j>>3)*512 + b*8 + (j&7)];
  else        v = P.ws[BM_OFF  + (long long)((j-24)>>3)*512 + b*8 + ((j-24)&7)];
  P.ws[WB_OFF + b*48 + j] = v;
}

// ================= zc[layer][b,n] = lb + rcb + wb . rcW[n,:] =================
__global__ __launch_bounds__(256) void k_zc(PtrPack P) {
  int idx = blockIdx.x*256 + threadIdx.x;          // 1536 total
  if (idx >= 1536) return;
  const int layer = idx / 512, r = idx % 512, b = r / 8, nn = r % 8;
  float a = P.lb[layer][nn] + P.rcb[layer][nn];
  const float* wbp = P.ws + WB_OFF + b*48;
  const float* rcW = P.rcW[layer] + nn*48;
  #pragma unroll 8
  for (int j = 0; j < 48; ++j) a += wbp[j] * rcW[j];
  P.ws[ZC_OFF + layer*512 + b*8 + nn] = a;
}

// ================= z3 vectors: (B,8,in_dim) per layer =================
__global__ __launch_bounds__(256) void k_z3(PtrPack P) {
  int idx = blockIdx.x*256 + threadIdx.x;          // 327680 total
  int layer, within;
  if (idx < 65536)       { layer = 0; within = idx; }
  else if (idx < 196608) { layer = 1; within = idx - 65536; }
  else                   { layer = 2; within = idx - 196608; }
  const int in = ndim_of(layer);
  const int b = within / (8*in);
  const int r = within % (8*in);
  const int nn = r / in, x = r % in;
  const int RI = (layer==0) ? 8 : 24;
  const float* rW = P.rW[layer] + nn*RI;
  const long long bo = (long long)b * 8;
  float a = P.rb[layer][nn];
  const float* rm = P.ws + rm_off(layer);
  #pragma unroll
  for (int c = 0; c < 8; ++c) a += rW[c] * rm[(bo+c)*in + x];
  if (layer > 0) {
    const float* cmp = P.ws + cm_off(layer-1);
    const float* bp  = P.bias[layer-1];
    #pragma unroll
    for (int c = 0; c < 8; ++c) a += rW[8+c]  * cmp[(bo+c)*256 + x];
    #pragma unroll
    for (int c = 0; c < 8; ++c) a += rW[16+c] * bp[(bo+c)*256 + x];
  }
  P.ws[z3_off(layer) + (bo+nn)*in + x] = a;
}

// ================= z4 vectors + bias outputs (both use the same "col" vector) =====
__global__ __launch_bounds__(256) void k_z4_outb(PtrPack P) {
  int idx = blockIdx.x*256 + threadIdx.x;          // 294912 total
  int layer, within;
  if (idx < 131072)      { layer = 0; within = idx; }
  else if (idx < 262144) { layer = 1; within = idx - 131072; }
  else                   { layer = 2; within = idx - 262144; }
  const int o = odim_of(layer);
  const int b = within / (8*o);
  const int r = within % (8*o);
  const int nn = r / o, y = r % o;
  const int CI = (layer==2) ? 16 : 24;
  const float* cW = P.cW[layer] + nn*CI;
  const float* bW = P.bW[layer] + nn*CI;
  const long long bo = (long long)b * 8;
  float a4 = P.cb[layer][nn];
  float ab = P.bb[layer][nn] + P.brcb[layer][nn];
  const float* cm = P.ws + cm_off(layer);
  #pragma unroll
  for (int c = 0; c < 8; ++c) {
    float v = cm[(bo+c)*o + y];
    a4 += cW[c]*v; ab += bW[c]*v;
  }
  const float* bi = P.bias[layer];
  #pragma unroll
  for (int c = 0; c < 8; ++c) {
    float v = bi[(bo+c)*o + y];
    a4 += cW[8+c]*v; ab += bW[8+c]*v;
  }
  if (layer < 2) {
    const float* rmn = P.ws + rm_off(layer+1);   // in_{l+1} == 256 == out_l
    #pragma unroll
    for (int c = 0; c < 8; ++c) {
      float v = rmn[(bo+c)*256 + y];
      a4 += cW[16+c]*v; ab += bW[16+c]*v;
    }
  }
  const float* wbp = P.ws + WB_OFF + b*48;
  const float* brcW = P.brcW[layer] + nn*48;
  #pragma unroll 8
  for (int j = 0; j < 48; ++j) ab += wbp[j] * brcW[j];
  P.ws[z4_off(layer) + (bo+nn)*o + y] = a4;
  P.out[ob_off(layer) + (bo+nn)*o + y] = ab;
}

// ================= Main fused pass: z1 via WMMA f32 + broadcast epilogue =========
// Each block: one (layer,b) plane chunk of 1024 flat positions, all 8 channels.
// D(16 pos x 16 ch) = A(16 pos x 4 ch) * B(4 ch x 16) accumulated twice (K=8).
#define LSTR 1032   // LDS channel stride (floats); 1032%64==8 avoids A-gather conflicts
__global__ __launch_bounds__(256) void k_main_wmma(PtrPack P) {
  __shared__ float lds[8 * LSTR];   // 33 KB
  const int blk = blockIdx.x;
  int layer, b, chunk;
  if (blk < 2048)      { layer = 0; b = blk >> 5;          chunk = blk & 31; }
  else if (blk < 6144) { layer = 1; b = (blk-2048) >> 6;   chunk = (blk-2048) & 63; }
  else                 { layer = 2; b = (blk-6144) >> 4;   chunk = (blk-6144) & 15; }
  const int odim  = odim_of(layer);
  const int nlog2 = (layer==0) ? 7 : 8;
  const int nd    = 1 << nlog2;
  const int Pn    = odim << nlog2;            // plane size o*n
  const long long p0 = (long long)chunk * 1024;
  const long long bo = (long long)b * 8;
  const float* wsrc = P.w[layer];
  const float* lW   = P.lW[layer];
  const int tid  = threadIdx.x;
  const int lane = tid & 31, wave = tid >> 5;
  const int half = lane >> 4, nn = lane & 15;

  // ---- stage 8 channel planes x 1024 positions into LDS ----
#if defined(HAVE_ASYNC_LDS)
  // CDNA5 async copy: global -> LDS without VGPR round-trip (ASYNCcnt-tracked)
  #pragma unroll
  for (int ch = 0; ch < 8; ++ch) {
    const float* g = wsrc + (bo + ch)*Pn + p0 + 4*tid;
    __builtin_amdgcn_global_load_async_to_lds_b128(
        (AS1 v4i*)g,
        (AS3 v4i*)&lds[ch*LSTR + 4*tid],
        0, 0);
  }
#if __has_builtin(__builtin_amdgcn_s_wait_asynccnt)
  __builtin_amdgcn_s_wait_asynccnt(0);
#else
  asm volatile("s_wait_asynccnt 0x0" ::: "memory");
#endif
#else
  #pragma unroll
  for (int ch = 0; ch < 8; ++ch) {
    float4 v = *(const float4*)(wsrc + (bo + ch)*Pn + p0 + 4*tid);
    *(float4*)&lds[ch*LSTR + 4*tid] = v;
  }
#endif

  // ---- B fragment: B[k][col] = lW[col][k] (lWT), cols 8..15 zero ----
  // layout: VGPR0 holds K = 2*half, VGPR1 holds K = 2*half+1 (matches A pattern)
  const int kb = 2*half;
  v2f B1, B2;
  if (nn < 8) {
    B1.x = lW[nn*8 + kb];     B1.y = lW[nn*8 + kb + 1];
    B2.x = lW[nn*8 + kb + 4]; B2.y = lW[nn*8 + kb + 5];
  } else { B1.x = 0.f; B1.y = 0.f; B2.x = 0.f; B2.y = 0.f; }

  __syncthreads();

  // ---- 8 WMMA tile-pairs per wave: 64 tiles x 16 positions = 1024 ----
  const int ch0 = 2*half;
  v8f acc[8];
  #pragma unroll
  for (int t = 0; t < 8; ++t) {
    const int toff = (wave*8 + t) * 16;
    const int pm = toff + nn;          // A fragment: lane holds position pm, channels per half
    v2f A1, A2;
    A1.x = lds[(ch0    )*LSTR + pm];
    A1.y = lds[(ch0 + 1)*LSTR + pm];
    A2.x = lds[(ch0 + 4)*LSTR + pm];
    A2.y = lds[(ch0 + 5)*LSTR + pm];
    v8f c = {0.f,0.f,0.f,0.f,0.f,0.f,0.f,0.f};
    c = __builtin_amdgcn_wmma_f32_16x16x4_f32(false, A1, false, B1, (short)0, c, false, false);
    c = __builtin_amdgcn_wmma_f32_16x16x4_f32(false, A2, false, B2, (short)0, c, false, false);
    acc[t] = c;
  }
  __syncthreads();

  // ---- scatter D fragments back to LDS as out[ch][pos] ----
  #pragma unroll
  for (int t = 0; t < 8; ++t) {
    const int toff = (wave*8 + t) * 16;
    if (nn < 8) {
      #pragma unroll
      for (int r = 0; r < 8; ++r)
        lds[nn*LSTR + toff + r + 8*half] = acc[t][r];   // D: row M=r+8*half, col N=nn
    }
  }
  __syncthreads();

  // ---- fused epilogue: + zc + z4[row] + z3[col], coalesced b128 stores ----
  const int q = 4*tid;
  const long long pos = p0 + q;
  const int x = (int)(pos >> nlog2);
  const int y = (int)(pos & (nd - 1));
  const float* zcp = P.ws + ZC_OFF + layer*512 + b*8;
  #pragma unroll
  for (int ch = 0; ch < 8; ++ch) {
    float4 v = *(float4*)&lds[ch*LSTR + q];
    float s = zcp[ch] + P.ws[z4_off(layer) + (bo+ch)*odim + x];
    float4 z = *(const float4*)(P.ws + z3_off(layer) + (bo+ch)*nd + y);
    v.x += s + z.x; v.y += s + z.y; v.z += s + z.z; v.w += s + z.w;
    *(float4*)(P.out + ow_off(layer) + (bo+ch)*Pn + pos) = v;
  }
}

extern "C" void kernel_launch(void* const* d_in, const int* in_sizes, int n_in,
                              void* d_out, int out_size, void* d_ws, size_t ws_size,
                              hipStream_t stream) {
  (void)in_sizes; (void)n_in; (void)out_size;
  if (ws_size < 1252864ull * sizeof(float)) return;   // need ~4.8 MB scratch

  PtrPack P;
  P.w[0]    = (const float*)d_in[0]; P.bias[0] = (const float*)d_in[1];
  P.w[1]    = (const float*)d_in[2]; P.bias[1] = (const float*)d_in[3];
  P.w[2]    = (const float*)d_in[4]; P.bias[2] = (const float*)d_in[5];
  for (int i = 0; i < 3; ++i) {
    const int base = 6 + 12*i;
    P.lW[i]   = (const float*)d_in[base + 0];
    P.lb[i]   = (const float*)d_in[base + 1];
    P.rcW[i]  = (const float*)d_in[base + 2];
    P.rcb[i]  = (const float*)d_in[base + 3];
    P.rW[i]   = (const float*)d_in[base + 4];
    P.rb[i]   = (const float*)d_in[base + 5];
    P.cW[i]   = (const float*)d_in[base + 6];
    P.cb[i]   = (const float*)d_in[base + 7];
    P.bW[i]   = (const float*)d_in[base + 8];
    P.bb[i]   = (const float*)d_in[base + 9];
    P.brcW[i] = (const float*)d_in[base + 10];
    P.brcb[i] = (const float*)d_in[base + 11];
  }
  P.ws  = (float*)d_ws;
  P.out = (float*)d_out;

  // Pass 1: reductions
  hipLaunchKernelGGL(k_reduce_w, dim3(1536), dim3(256), 0, stream, P);
  hipLaunchKernelGGL(k_reduce_b, dim3(192),  dim3(256), 0, stream, P);
  // Small vector precomputes (ordered on the stream)
  hipLaunchKernelGGL(k_wb,      dim3(12),   dim3(256), 0, stream, P);
  hipLaunchKernelGGL(k_zc,      dim3(6),    dim3(256), 0, stream, P);
  hipLaunchKernelGGL(k_z3,      dim3(1280), dim3(256), 0, stream, P);
  hipLaunchKernelGGL(k_z4_outb, dim3(1152), dim3(256), 0, stream, P);
  // Pass 2: fused WMMA channel-mix + broadcast epilogue
  hipLaunchKernelGGL(k_main_wmma, dim3(7168), dim3(256), 0, stream, P);
}